// Net_orig_32229434589355
// MI455X (gfx1250) — compile-verified
//
#include <hip/hip_runtime.h>
#include <math.h>

typedef __attribute__((ext_vector_type(2))) float v2f;
typedef __attribute__((ext_vector_type(8))) float v8f;

#define N_IN 128
#define N_HID 64
#define N_OUT 40
#define N_OUT_PAD 48

// ---------------- degree / normalization ----------------

__global__ void k_deg_init(float* __restrict__ deg, int n) {
  int i = blockIdx.x * blockDim.x + threadIdx.x;
  if (i < n) deg[i] = 1.0f;  // self-loop
}

__global__ void k_deg_edges(const int* __restrict__ dst, float* __restrict__ deg, int e) {
  int i = blockIdx.x * blockDim.x + threadIdx.x;
  if (i < e)
    __hip_atomic_fetch_add(&deg[dst[i]], 1.0f, __ATOMIC_RELAXED, __HIP_MEMORY_SCOPE_AGENT);
}

__global__ void k_dinv(float* __restrict__ deg, int n) {
  int i = blockIdx.x * blockDim.x + threadIdx.x;
  if (i < n) deg[i] = rsqrtf(deg[i]);  // deg >= 1 always
}

// ---------------- GEMM1: H1[n,64] = X[n,128] @ W1[128,64] (fp32 WMMA) ----------------
// Block = 256 thr = 8 waves; block tile = 32 rows x 64 cols (2x4 wave tiles of 16x16).

__global__ __launch_bounds__(256) void k_gemm1(const float* __restrict__ X,
                                               const float* __restrict__ W,
                                               float* __restrict__ H, int n) {
  __shared__ float Ws[N_IN * 68];   // 128x64, rows padded to 68 floats (bank spread)
  __shared__ float Xs[32 * 132];    // 32x128, rows padded to 132 floats
  const int tid = threadIdx.x;
  const int row_base = blockIdx.x * 32;
  const bool full = (row_base + 32 <= n);   // block-uniform: no per-row guards needed

  for (int idx = tid; idx < N_IN * N_HID; idx += 256) {
    int r = idx >> 6, c = idx & 63;
    Ws[r * 68 + c] = W[idx];
  }
  if (full) {
    for (int idx = tid; idx < 32 * N_IN; idx += 256) {
      int r = idx >> 7, c = idx & 127;
      Xs[r * 132 + c] = X[(size_t)(row_base + r) * N_IN + c];
    }
  } else {
    for (int idx = tid; idx < 32 * N_IN; idx += 256) {
      int r = idx >> 7, c = idx & 127;
      int row = row_base + r;
      Xs[r * 132 + c] = (row < n) ? X[(size_t)row * N_IN + c] : 0.0f;
    }
  }
  __syncthreads();

  const int w    = tid >> 5;           // wave 0..7
  const int lane = tid & 31;
  const int rt = w >> 2, ct = w & 3;   // 2 row tiles x 4 col tiles
  const int m  = lane & 15;
  const int kq = (lane >> 4) << 1;     // lanes 0-15: K+0/K+1 ; lanes 16-31: K+2/K+3
  const int r0 = rt * 16, c0 = ct * 16;

  v8f acc = {};
  const float* xrow = &Xs[(r0 + m) * 132];
  #pragma unroll 4
  for (int k = 0; k < N_IN; k += 4) {
    v2f a, b;
    a.x = xrow[k + kq];
    a.y = xrow[k + kq + 1];
    b.x = Ws[(k + kq) * 68 + c0 + m];
    b.y = Ws[(k + kq + 1) * 68 + c0 + m];
    acc = __builtin_amdgcn_wmma_f32_16x16x4_f32(false, a, false, b, (short)0, acc,
                                                false, false);
  }
  const int rowoff = (lane >> 4) << 3;  // C/D layout: lanes 16-31 hold M = r+8
  float* hp = &H[(size_t)(row_base + r0 + rowoff) * N_HID + c0 + m];
  if (full) {
    #pragma unroll
    for (int r = 0; r < 8; ++r) hp[(size_t)r * N_HID] = acc[r];
  } else {
    #pragma unroll
    for (int r = 0; r < 8; ++r) {
      int row = row_base + r0 + r + rowoff;
      if (row < n) H[(size_t)row * N_HID + c0 + m] = acc[r];
    }
  }
}

// ---------------- GEMM2: H2[n,40] = A[n,64] @ W2[64,40] (pad N to 48) ----------------
// Block = 192 thr = 6 waves; block tile = 32 rows x 48 cols (2x3 wave tiles).

__global__ __launch_bounds__(192) void k_gemm2(const float* __restrict__ A,
                                               const float* __restrict__ W2,
                                               float* __restrict__ H, int n) {
  __shared__ float Ws[N_HID * 52];  // 64x48 padded to 52
  __shared__ float As[32 * 68];     // 32x64 padded to 68
  const int tid = threadIdx.x;
  const int row_base = blockIdx.x * 32;
  const bool full = (row_base + 32 <= n);

  for (int idx = tid; idx < N_HID * N_OUT_PAD; idx += 192) {
    int r = idx / N_OUT_PAD, c = idx % N_OUT_PAD;
    Ws[r * 52 + c] = (c < N_OUT) ? W2[r * N_OUT + c] : 0.0f;
  }
  if (full) {
    for (int idx = tid; idx < 32 * N_HID; idx += 192) {
      int r = idx >> 6, c = idx & 63;
      As[r * 68 + c] = A[(size_t)(row_base + r) * N_HID + c];
    }
  } else {
    for (int idx = tid; idx < 32 * N_HID; idx += 192) {
      int r = idx >> 6, c = idx & 63;
      int row = row_base + r;
      As[r * 68 + c] = (row < n) ? A[(size_t)row * N_HID + c] : 0.0f;
    }
  }
  __syncthreads();

  const int w    = tid >> 5;          // wave 0..5
  const int lane = tid & 31;
  const int rt = w / 3, ct = w % 3;
  const int m  = lane & 15;
  const int kq = (lane >> 4) << 1;
  const int r0 = rt * 16, c0 = ct * 16;

  v8f acc = {};
  const float* arow = &As[(r0 + m) * 68];
  #pragma unroll 4
  for (int k = 0; k < N_HID; k += 4) {
    v2f a, b;
    a.x = arow[k + kq];
    a.y = arow[k + kq + 1];
    b.x = Ws[(k + kq) * 52 + c0 + m];
    b.y = Ws[(k + kq + 1) * 52 + c0 + m];
    acc = __builtin_amdgcn_wmma_f32_16x16x4_f32(false, a, false, b, (short)0, acc,
                                                false, false);
  }
  const int rowoff = (lane >> 4) << 3;
  const int col = c0 + m;
  if (col < N_OUT) {
    if (full) {
      float* hp = &H[(size_t)(row_base + r0 + rowoff) * N_OUT + col];
      #pragma unroll
      for (int r = 0; r < 8; ++r) hp[(size_t)r * N_OUT] = acc[r];
    } else {
      #pragma unroll
      for (int r = 0; r < 8; ++r) {
        int row = row_base + r0 + r + rowoff;
        if (row < n) H[(size_t)row * N_OUT + col] = acc[r];
      }
    }
  }
}

// ---------------- aggregation (F is compile-time: cheap div/mod, 32-bit idx) ----------

// out[i,f] = dinv[i]^2 * h[i,f]   (self-loop contribution; also zero-inits buffer)
template <int F>
__global__ void k_agg_init(const float* __restrict__ h, const float* __restrict__ dinv,
                           float* __restrict__ out, unsigned total) {
  unsigned idx = blockIdx.x * blockDim.x + threadIdx.x;
  if (idx < total) {
    unsigned i = idx / F;
    float di = dinv[i];
    out[idx] = di * di * h[idx];
  }
}

// out[dst,f] += dinv[src]*dinv[dst]*h[src,f]; consecutive threads cover one row -> coalesced
template <int F>
__global__ void k_agg_edges(const float* __restrict__ h, const float* __restrict__ dinv,
                            const int* __restrict__ src, const int* __restrict__ dst,
                            float* __restrict__ out, unsigned total) {
  unsigned idx = blockIdx.x * blockDim.x + threadIdx.x;
  if (idx >= total) return;
  unsigned e = idx / F;
  unsigned f = idx - e * F;
  int s = src[e], d = dst[e];
  float wgt = dinv[s] * dinv[d];
  __hip_atomic_fetch_add(&out[(unsigned)d * F + f], wgt * h[(unsigned)s * F + f],
                         __ATOMIC_RELAXED, __HIP_MEMORY_SCOPE_AGENT);
}

template <int F>
__global__ void k_bias_relu(float* __restrict__ a, const float* __restrict__ b,
                            unsigned total) {
  unsigned idx = blockIdx.x * blockDim.x + threadIdx.x;
  if (idx < total) {
    unsigned f = idx % F;
    a[idx] = fmaxf(a[idx] + b[f], 0.0f);
  }
}

// per-node: add bias, numerically-stable log_softmax over 40 classes, in place
__global__ void k_bias_logsoftmax(float* __restrict__ out, const float* __restrict__ b2,
                                  int n) {
  int i = blockIdx.x * blockDim.x + threadIdx.x;
  if (i >= n) return;
  float v[N_OUT];
  float mx = -INFINITY;
  #pragma unroll
  for (int f = 0; f < N_OUT; ++f) {
    v[f] = out[(size_t)i * N_OUT + f] + b2[f];
    mx = fmaxf(mx, v[f]);
  }
  float s = 0.0f;
  #pragma unroll
  for (int f = 0; f < N_OUT; ++f) s += expf(v[f] - mx);
  float lse = mx + logf(s);
  #pragma unroll
  for (int f = 0; f < N_OUT; ++f) out[(size_t)i * N_OUT + f] = v[f] - lse;
}

// ---------------- launcher ----------------

extern "C" void kernel_launch(void* const* d_in, const int* in_sizes, int n_in,
                              void* d_out, int out_size, void* d_ws, size_t ws_size,
                              hipStream_t stream) {
  const float* X  = (const float*)d_in[0];
  const int*   ei = (const int*)d_in[1];
  const float* W1 = (const float*)d_in[2];
  const float* b1 = (const float*)d_in[3];
  const float* W2 = (const float*)d_in[4];
  const float* b2 = (const float*)d_in[5];

  const int n = in_sizes[0] / N_IN;   // 100000
  const int e = in_sizes[1] / 2;      // 1600000
  const int* src = ei;                // edge_index[0]
  const int* dst = ei + e;            // edge_index[1]

  float* ws   = (float*)d_ws;
  float* dinv = ws;                                  // n
  float* H1   = dinv + n;                            // n*64
  float* A1   = H1 + (size_t)n * N_HID;              // n*64
  float* H2   = A1 + (size_t)n * N_HID;              // n*40
  float* OUT  = (float*)d_out;                       // n*40

  // degrees -> dinv
  k_deg_init<<<(n + 255) / 256, 256, 0, stream>>>(dinv, n);
  k_deg_edges<<<(e + 255) / 256, 256, 0, stream>>>(dst, dinv, e);
  k_dinv<<<(n + 255) / 256, 256, 0, stream>>>(dinv, n);

  const int gblocks = (n + 31) / 32;

  // layer 1
  k_gemm1<<<gblocks, 256, 0, stream>>>(X, W1, H1, n);
  unsigned t1  = (unsigned)n * N_HID;     // 6.4M
  unsigned te1 = (unsigned)e * N_HID;     // 102.4M  (< 2^31)
  k_agg_init<N_HID><<<(t1 + 255) / 256, 256, 0, stream>>>(H1, dinv, A1, t1);
  k_agg_edges<N_HID><<<(te1 + 255) / 256, 256, 0, stream>>>(H1, dinv, src, dst, A1, te1);
  k_bias_relu<N_HID><<<(t1 + 255) / 256, 256, 0, stream>>>(A1, b1, t1);

  // layer 2
  k_gemm2<<<gblocks, 192, 0, stream>>>(A1, W2, H2, n);
  unsigned t2  = (unsigned)n * N_OUT;     // 4M
  unsigned te2 = (unsigned)e * N_OUT;     // 64M
  k_agg_init<N_OUT><<<(t2 + 255) / 256, 256, 0, stream>>>(H2, dinv, OUT, t2);
  k_agg_edges<N_OUT><<<(te2 + 255) / 256, 256, 0, stream>>>(H2, dinv, src, dst, OUT, te2);
  k_bias_logsoftmax<<<(n + 255) / 256, 256, 0, stream>>>(OUT, b2, n);
}